// RNNTLoss_58600533786917
// MI455X (gfx1250) — compile-verified
//
#include <hip/hip_runtime.h>
#include <hip/hip_bf16.h>
#include <math.h>

// ---------------- problem constants (match reference) ----------------------
constexpr int BB = 16;
constexpr int TT = 256;
constexpr int UU = 64;
constexpr int VV = 128;
constexpr int U1 = UU + 1;          // 65 columns (u = 0..64)
constexpr int SBP = U1 + 1;         // 66: padded blank-row stride -> conflict-free
constexpr float NEG = -1e30f;
constexpr int BLOCK = 128;          // 4 wave32s for staging; wave 0 runs the DP
constexpr int SMEM_FLOATS = TT * SBP + TT * UU + UU;

// ------------- async global -> LDS copy (CDNA5 ASYNCcnt path) --------------
typedef __attribute__((address_space(1))) int glob_i32;
typedef __attribute__((address_space(3))) int lds_i32;

__device__ __forceinline__ void async_cp_f32(const float* g, float* l) {
  __builtin_amdgcn_global_load_async_to_lds_b32(
      (glob_i32*)const_cast<float*>(g), (lds_i32*)l, /*offset=*/0, /*cpol=*/0);
}

__device__ __forceinline__ void async_wait_all() {
  __builtin_amdgcn_s_wait_asynccnt(0);
}

// LSE(a,b) with -1e30 acting as -inf (matches reference surrogate semantics)
__device__ __forceinline__ float lse2(float a, float b) {
  float m = fmaxf(a, b);
  float d = fminf(a, b) - m;                 // <= 0
  return m + log1pf(__expf(d));
}

// ---------------------------- RNNT forward DP ------------------------------
// One workgroup per batch element.
// Phase 1 (all 4 waves): async-gather blank[T][66] and emit[T][64] planes
//   into LDS (ASYNCcnt-tracked, no VGPR round trip), then s_wait_asynccnt+bar.
// Phase 2 (wave 0 only): anti-diagonal wavefront, lane l owns columns
//   u=2l, 2l+1 (lane 31 also u=64). Cross-lane dependency is one __shfl_up
//   per step -> zero barriers, zero comm-LDS on the 320-step critical path.
__global__ __launch_bounds__(BLOCK) void rnnt_loss_kernel(
    const float* __restrict__ x,   // (B, T, U+1, V) log-probs
    const int*   __restrict__ y,   // (B, U) labels
    const int*   __restrict__ xl,  // (B,)
    const int*   __restrict__ yl,  // (B,)
    float*       __restrict__ out) // (B,) negative log-likelihood
{
  extern __shared__ float smem[];
  float* sb = smem;                        // blank plane [T][SBP] (padded)
  float* se = sb + TT * SBP;               // emit  plane [T][UU]
  int*   ly = (int*)(se + TT * UU);        // labels [UU]

  const int tid = threadIdx.x;
  const int b   = blockIdx.x;
  const float* xb = x + (size_t)b * TT * U1 * VV;

  if (tid < UU) ly[tid] = y[b * UU + tid];
  __syncthreads();

  // ---- stage blank plane: sb[t*SBP+u] = x[b,t,u,0]  (16640 elements)
  for (int i = tid; i < TT * U1; i += BLOCK) {
    int t = i / U1;
    int u = i - t * U1;
    async_cp_f32(xb + (size_t)(t * U1 + u) * VV, sb + t * SBP + u);
  }
  // ---- stage emit plane: se[t*UU+u] = x[b,t,u,y[b,u]] (16384 elements)
  {
    const int u   = tid & (UU - 1);          // loop-invariant per thread
    const int col = ly[u];                   // hoisted label read
    for (int t = tid >> 6; t < TT; t += BLOCK / UU) {
      async_cp_f32(xb + (size_t)(t * U1 + u) * VV + col, se + t * UU + u);
    }
  }
  async_wait_all();
  __syncthreads();

  if (tid >= 32) return;                     // DP runs on wave 0 only
  const int l = tid;                         // lane id
  const int xlb = max(1, min(xl[b], TT));    // reference clips xl to [1, T]
  const int ylb = yl[b];
  const int u0 = 2 * l, u1 = 2 * l + 1;      // owned columns (u2 = 64 on lane 31)

  float a0 = NEG, a1 = NEG, a2 = NEG;        // alpha at owned columns, prev step
  float res = 0.0f; bool hit = false;

  // s = t + u anti-diagonal; max needed s = (T-1) + U = 319
  for (int s = 0; s < TT + UU; ++s) {
    const float oldA0 = a0, oldA1 = a1, oldA2 = a2;
    const float leftIn = __shfl_up(oldA1, 1, 32);   // lane l-1's alpha(t0, u0-1)

    // ---- cell 0: (t0, u0 = 2l) ----
    const int t0 = s - u0;
    if (t0 >= 0 && t0 < TT) {
      float up   = (t0 > 0) ? oldA0 + sb[(t0 - 1) * SBP + u0] : NEG;
      float left = (l  > 0) ? leftIn + se[t0 * UU + (u0 - 1)] : NEG;
      float a = lse2(up, left);
      if (t0 == 0 && l == 0) a = 0.0f;              // alpha(0,0) = 0
      a0 = a;
      if (t0 == xlb - 1 && u0 == ylb) { res = -(a + sb[t0 * SBP + u0]); hit = true; }
    }
    // ---- cell 1: (t1, u1 = 2l+1); left neighbor is own cell 0 (prev step) ----
    const int t1 = t0 - 1;
    if (t1 >= 0 && t1 < TT) {
      float up   = (t1 > 0) ? oldA1 + sb[(t1 - 1) * SBP + u1] : NEG;
      float left = oldA0 + se[t1 * UU + (u1 - 1)];
      float a = lse2(up, left);
      a1 = a;
      if (t1 == xlb - 1 && u1 == ylb) { res = -(a + sb[t1 * SBP + u1]); hit = true; }
    }
    // ---- cell 2: (t2, u=64) on lane 31; left neighbor is own cell 1 ----
    const int t2 = s - UU;                          // uniform across lanes
    if (t2 >= 0 && t2 < TT) {
      float up   = (t2 > 0) ? oldA2 + sb[(t2 - 1) * SBP + UU] : NEG;
      float left = oldA1 + se[t2 * UU + (UU - 1)];
      float a = lse2(up, left);
      a2 = a;
      if (l == 31 && t2 == xlb - 1 && UU == ylb) { res = -(a + sb[t2 * SBP + UU]); hit = true; }
    }
  }
  if (hit) out[b] = res;                            // exactly one lane hits
}

// ------------------------------- launcher ----------------------------------
extern "C" void kernel_launch(void* const* d_in, const int* in_sizes, int n_in,
                              void* d_out, int out_size, void* d_ws, size_t ws_size,
                              hipStream_t stream) {
  const float* x  = (const float*)d_in[0];
  const int*   y  = (const int*)d_in[1];
  const int*   xl = (const int*)d_in[2];
  const int*   yl = (const int*)d_in[3];
  float* out = (float*)d_out;
  (void)in_sizes; (void)n_in; (void)out_size; (void)d_ws; (void)ws_size;

  const size_t smem_bytes = (size_t)SMEM_FLOATS * sizeof(float);  // ~133 KB, fits 320 KB/WGP
  rnnt_loss_kernel<<<BB, BLOCK, smem_bytes, stream>>>(x, y, xl, yl, out);
}